// NMSLoss3_87136296501784
// MI455X (gfx1250) — compile-verified
//
#include <hip/hip_runtime.h>
#include <math.h>

#define NMS_THR 0.5f
#define EPS_F   1e-6f

constexpr int N_  = 2048;   // proposals per batch
constexpr int G_  = 64;     // gt boxes per batch
constexpr int NT  = 1024;   // threads per block (32 wave32 waves)
constexpr int NW  = NT / 32;
constexpr int PER = N_ / NT;

typedef __attribute__((ext_vector_type(4))) unsigned int v4u;
typedef __attribute__((ext_vector_type(8))) int v8i;
typedef __attribute__((ext_vector_type(4))) int v4i;

__device__ inline float iou_pair(const float* a, const float* b) {
    float areaA = (a[2] - a[0]) * (a[3] - a[1]);
    float areaB = (b[2] - b[0]) * (b[3] - b[1]);
    float lx = fmaxf(a[0], b[0]);
    float ly = fmaxf(a[1], b[1]);
    float rx = fminf(a[2], b[2]);
    float ry = fminf(a[3], b[3]);
    float w = fmaxf(rx - lx, 0.f), h = fmaxf(ry - ly, 0.f);
    float inter = w * h;
    return inter / (areaA + areaB - inter + 1e-12f);
}

// Issue a 1-D TDM transfer global -> LDS. nelem = number of 4-byte elements
// (must fit in tile_dim0's 16 bits). Descriptor layout per CDNA5 ISA sec. 8.3/8.4.
// Toolchain here is the 6-arg builtin: (v4u g0, v8i g1, v4i g2, v4i g3, v8i, i32 cpol).
__device__ inline void tdm_load_1d(unsigned lds_off, const void* gptr, unsigned nelem) {
    unsigned long long ga = (unsigned long long)gptr;
    v4u g0;
    g0.x = 1u;                                   // count=1 (valid), is_restore=0, no gather
    g0.y = lds_off;                              // lds_addr (bytes)
    g0.z = (unsigned)(ga & 0xFFFFFFFFu);         // global_addr[31:0]
    g0.w = (unsigned)((ga >> 32) & 0x1FFFFFFu)   // global_addr[56:32]
         | (2u << 30);                           // type = 2 ("image")
    v8i g1;
    g1[0] = (int)(2u << 16);                     // data_size = 2 (4 bytes); wg_mask=0
    g1[1] = (int)((nelem & 0xFFFFu) << 16);      // tensor_dim0[15:0]  @ bits 63:48
    g1[2] = (int)((nelem >> 16) | (1u << 16));   // tensor_dim0[31:16]; tensor_dim1 = 1
    g1[3] = (int)((nelem & 0xFFFFu) << 16);      // tile_dim0 = nelem  @ bits 127:112
    g1[4] = 0;                                   // tile_dim1 = tile_dim2 = 0 (unused)
    g1[5] = (int)nelem;                          // tensor_dim0_stride[31:0]
    g1[6] = 0;
    g1[7] = 0;
    v4i z4 = {0, 0, 0, 0};
    v8i z8 = {0, 0, 0, 0, 0, 0, 0, 0};
    __builtin_amdgcn_tensor_load_to_lds(g0, g1, z4, z4, z8, 0);
}

__global__ __launch_bounds__(NT) void nms_loss_kernel(
    const int* __restrict__ gi_g,      // anchor_gt_inds  (B,N) int32
    const float* __restrict__ gtb_g,   // gt_bboxes       (B,G,4) f32
    const float* __restrict__ prop_g,  // proposal_list   (B,N,5) f32
    float* __restrict__ out, int B)
{
    __shared__ float prop[N_ * 5];      // boxes + score, stride 5 (gcd(5,64)=1: bank-conflict free)
    __shared__ int   gis[N_];
    __shared__ float gtb[G_ * 4];
    __shared__ float gtiou[G_ * G_];
    __shared__ unsigned char alive[N_];
    __shared__ int   rec[G_];
    __shared__ float rv[NW];  __shared__ int ri[NW];        // argmax partials
    __shared__ float rs[NW];  __shared__ float rc[NW];      // push partials
    __shared__ int   rk[NW];                                // kill-count partials
    __shared__ int   bc_i, bc_g, bc_done;

    const int b = blockIdx.x;
    const int tid = threadIdx.x;
    const int wid = tid >> 5, lane = tid & 31;

    // ---- fill LDS with this batch's tensors via the Tensor Data Mover ----
    if (tid < 32) {
        tdm_load_1d((unsigned)(size_t)(prop), prop_g + (size_t)b * N_ * 5, N_ * 5);
        tdm_load_1d((unsigned)(size_t)(gis),  gi_g  + (size_t)b * N_,      N_);
        tdm_load_1d((unsigned)(size_t)(gtb),  gtb_g + (size_t)b * G_ * 4,  G_ * 4);
        __builtin_amdgcn_s_wait_tensorcnt(0);
    }
    for (int j = tid; j < N_; j += NT) alive[j] = 1;
    if (tid < G_) rec[tid] = -1;
    if (tid == 0) bc_done = 0;
    __syncthreads();

    // ---- gt x gt IoU table (64x64) ----
    for (int t = tid; t < G_ * G_; t += NT) {
        int r = t >> 6, c = t & 63;
        gtiou[t] = iou_pair(&gtb[r * 4], &gtb[c * 4]);
    }
    __syncthreads();

    float pull_s = 0.f, pull_c = 0.f, push_s = 0.f, push_c = 0.f;  // thread0 only
    int alivecnt = N_;                                             // thread0 only
    bool pending = false;                                          // thread0 only

    for (;;) {
        // thread0: consume previous step's suppression partials (after last barrier)
        if (tid == 0 && pending) {
            float psum = 0.f, cntf = 0.f; int kc = 0;
            for (int w = 0; w < NW; ++w) { psum += rs[w]; cntf += rc[w]; kc += rk[w]; }
            float push = (cntf > 0.f) ? (psum / cntf) : 0.f;
            push_s += push; push_c += cntf;
            alivecnt -= kc;
        }
        // ---- argmax over alive scores ----
        float bv = -1.f; int bi = -1;
        #pragma unroll
        for (int k = 0; k < PER; ++k) {
            int j = tid + k * NT;
            if (alive[j]) { float s = prop[j * 5 + 4]; if (s > bv) { bv = s; bi = j; } }
        }
        for (int off = 16; off; off >>= 1) {
            float ov = __shfl_down(bv, off);
            int   oi = __shfl_down(bi, off);
            if (ov > bv || (ov == bv && (unsigned)oi < (unsigned)bi)) { bv = ov; bi = oi; }
        }
        if (lane == 0) { rv[wid] = bv; ri[wid] = bi; }
        __syncthreads();

        if (tid < 32) {
            bv = rv[lane]; bi = ri[lane];
            for (int off = 16; off; off >>= 1) {
                float ov = __shfl_down(bv, off);
                int   oi = __shfl_down(bi, off);
                if (ov > bv || (ov == bv && (unsigned)oi < (unsigned)bi)) { bv = ov; bi = oi; }
            }
            if (tid == 0) {
                if (bi < 0) {
                    bc_done = 1;
                } else {
                    int i = bi, g = gis[i];
                    int rg = rec[g];
                    bool has = rg >= 0;
                    int mi = has ? rg : 0;
                    float msiou = fmaxf(iou_pair(&prop[mi * 5], &prop[i * 5]), EPS_F);
                    float pull  = -logf(1.f - NMS_THR + msiou) * prop[i * 5 + 4];
                    if (!has) rec[g] = i;
                    alive[i] = 0;
                    alivecnt -= 1;
                    bool remaining = alivecnt > 0;
                    if (has && remaining) pull_s += pull;
                    if (has) pull_c += 1.f;
                    bc_i = i; bc_g = g;
                }
            }
        }
        __syncthreads();
        if (bc_done) break;

        // ---- suppression + push terms ----
        const int i = bc_i, g = bc_g;
        const float bx0 = prop[i * 5 + 0], by0 = prop[i * 5 + 1];
        const float bx1 = prop[i * 5 + 2], by1 = prop[i * 5 + 3];
        const float areaI = (bx1 - bx0) * (by1 - by0);
        float psum = 0.f, cntf = 0.f; int kc = 0;
        #pragma unroll
        for (int k = 0; k < PER; ++k) {
            int j = tid + k * NT;
            if (alive[j]) {
                float a0 = prop[j * 5], a1 = prop[j * 5 + 1];
                float a2 = prop[j * 5 + 2], a3 = prop[j * 5 + 3];
                float areaJ = (a2 - a0) * (a3 - a1);
                float lx = fmaxf(bx0, a0), ly = fmaxf(by0, a1);
                float rx = fminf(bx1, a2), ry = fminf(by1, a3);
                float w = fmaxf(rx - lx, 0.f), h = fmaxf(ry - ly, 0.f);
                float inter = w * h;
                float cur = inter / (areaI + areaJ - inter + 1e-12f);
                if (cur > NMS_THR) {
                    kc += 1;
                    alive[j] = 0;
                    int gj = gis[j];
                    if (gj != g && cur > gtiou[g * G_ + gj]) {
                        float sc = prop[j * 5 + 4];
                        psum += (-logf(1.f + NMS_THR - cur) - logf(sc)) * sc;
                        cntf += 1.f;
                    }
                }
            }
        }
        for (int off = 16; off; off >>= 1) {
            psum += __shfl_down(psum, off);
            cntf += __shfl_down(cntf, off);
            kc   += __shfl_down(kc, off);
        }
        if (lane == 0) { rs[wid] = psum; rc[wid] = cntf; rk[wid] = kc; }
        pending = true;
        __syncthreads();
    }

    if (tid == 0) {
        float push_b = push_s / (push_c + EPS_F);
        float pull_b = pull_s / (pull_c + EPS_F);
        // PUSH_W = PULL_W = 1.0
        atomicAdd(&out[0], push_b / (float)B);
        atomicAdd(&out[1], pull_b / (float)B);
    }
}

__global__ void zero_out_kernel(float* out) {
    if (threadIdx.x < 2) out[threadIdx.x] = 0.f;
}

extern "C" void kernel_launch(void* const* d_in, const int* in_sizes, int n_in,
                              void* d_out, int out_size, void* d_ws, size_t ws_size,
                              hipStream_t stream) {
    // inputs: 0 gt_inds (B,N) i32 ; 1 anchor_gt_inds (B,N) i32 ;
    //         2 gt_bboxes (B,G,4) f32 ; 3 proposal_list (B,N,5) f32
    const int*   gi   = (const int*)d_in[1];
    const float* gtb  = (const float*)d_in[2];
    const float* prop = (const float*)d_in[3];
    float* out = (float*)d_out;
    const int B = in_sizes[1] / N_;

    zero_out_kernel<<<1, 32, 0, stream>>>(out);
    nms_loss_kernel<<<B, NT, 0, stream>>>(gi, gtb, prop, out, B);
}